// GNNModule_55576876810816
// MI455X (gfx1250) — compile-verified
//
#include <hip/hip_runtime.h>
#include <hip/hip_bf16.h>

typedef __attribute__((ext_vector_type(2))) float v2f;
typedef __attribute__((ext_vector_type(8))) float v8f;

#define HDIM 64

// ---------------------------------------------------------------- utilities
__global__ void zero_f32(float* __restrict__ p, int n) {
  int i = blockIdx.x * blockDim.x + threadIdx.x;
  if (i < n) p[i] = 0.0f;
}

__global__ void deg_accum(const int* __restrict__ dst, float* __restrict__ deg, int E) {
  int e = blockIdx.x * blockDim.x + threadIdx.x;
  if (e < E) atomicAdd(&deg[dst[e]], 1.0f);
}

__global__ void deg_to_dis(float* __restrict__ deg, int n) {
  int i = blockIdx.x * blockDim.x + threadIdx.x;
  if (i < n) {
    float d = deg[i];
    deg[i] = (d > 0.0f) ? rsqrtf(fmaxf(d, 1.0f)) : 0.0f;
  }
}

// ------------------------------------------------- WMMA GEMM: [n,K] x [K,64]
// One wave computes a 16x64 tile via 4 accumulators of V_WMMA_F32_16X16X4_F32.
// n_rows must be a multiple of 16 (N=100000 = 6250*16).
// f32 16x16x4 fragment layouts (ISA 7.12.2):
//   A (16x4): lanes 0-15 -> K=kb+{0,1}; lanes 16-31 -> K=kb+{2,3}; M = lane%16
//   B (4x16): symmetric striping over lanes; N = lane%16
//   D (16x16): VGPR r -> M = (lane/16)*8 + r, N = lane%16
template<int K, bool RELU_BIAS>
__global__ void gemm_n64_wmma(const float* __restrict__ A,
                              const float* __restrict__ W,
                              const float* __restrict__ bias,
                              float* __restrict__ out, int n_rows) {
  int wave = (blockIdx.x * blockDim.x + threadIdx.x) >> 5;
  int lane = threadIdx.x & 31;
  int row0 = wave * 16;
  if (row0 >= n_rows) return;              // whole-wave guard: EXEC stays all-ones

  const int half = lane >> 4;              // 0: K pair {0,1}; 1: K pair {2,3}
  const int l16  = lane & 15;

  v8f acc0 = {}, acc1 = {}, acc2 = {}, acc3 = {};

  const float* arow = A + (size_t)(row0 + l16) * K;
  for (int kb = 0; kb < K; kb += 4) {
    v2f a;
    a.x = arow[kb + half * 2 + 0];
    a.y = arow[kb + half * 2 + 1];

    const float* bp = W + (size_t)(kb + half * 2) * HDIM + l16;
    v2f b0, b1, b2, b3;
    b0.x = bp[0];        b0.y = bp[HDIM];
    b1.x = bp[16];       b1.y = bp[HDIM + 16];
    b2.x = bp[32];       b2.y = bp[HDIM + 32];
    b3.x = bp[48];       b3.y = bp[HDIM + 48];

    acc0 = __builtin_amdgcn_wmma_f32_16x16x4_f32(false, a, false, b0, (short)0, acc0, false, false);
    acc1 = __builtin_amdgcn_wmma_f32_16x16x4_f32(false, a, false, b1, (short)0, acc1, false, false);
    acc2 = __builtin_amdgcn_wmma_f32_16x16x4_f32(false, a, false, b2, (short)0, acc2, false, false);
    acc3 = __builtin_amdgcn_wmma_f32_16x16x4_f32(false, a, false, b3, (short)0, acc3, false, false);
  }

  v8f accs[4] = {acc0, acc1, acc2, acc3};
  #pragma unroll
  for (int nt = 0; nt < 4; ++nt) {
    int col = nt * 16 + l16;
    float bcol = RELU_BIAS ? bias[col] : 0.0f;
    #pragma unroll
    for (int r = 0; r < 8; ++r) {
      int m = half * 8 + r;
      float v = accs[nt][r];
      if (RELU_BIAS) { v += bcol; v = fmaxf(v, 0.0f); }
      out[(size_t)(row0 + m) * HDIM + col] = v;
    }
  }
}

// ----------------------------------------------- edge-parallel atomic scatter
// One wave per edge; each lane handles 2 columns (float2 gather, 2x f32 atomic
// add). hw + acc (~52MB) stay resident in the 192MB L2, so these atomics never
// hit HBM.
__global__ void scatter_edges(const int* __restrict__ src, const int* __restrict__ dst,
                              const float* __restrict__ dis,
                              const float* __restrict__ hw,
                              float* __restrict__ acc, int E) {
  int gid  = blockIdx.x * blockDim.x + threadIdx.x;
  int e    = gid >> 5;
  int lane = gid & 31;
  if (e >= E) return;
  int s = src[e];
  int d = dst[e];
  float coef = dis[s] * dis[d];
  const float2* hp = (const float2*)(hw + (size_t)s * HDIM);
  float2 v = hp[lane];
  float* ap = acc + (size_t)d * HDIM + lane * 2;
  atomicAdd(ap + 0, v.x * coef);
  atomicAdd(ap + 1, v.y * coef);
}

// ------------------------------------- h = relu(acc + b) + h   (skip + bias)
__global__ void finalize_layer(const float* __restrict__ acc,
                               const float* __restrict__ bias,
                               float* __restrict__ h, int n64) {
  int i = blockIdx.x * blockDim.x + threadIdx.x;
  if (i >= n64) return;
  float v = acc[i] + bias[i & (HDIM - 1)];
  h[i] = fmaxf(v, 0.0f) + h[i];
}

// ------------------------------------------------- head: out = h @ W_head + b
__global__ void head_kernel(const float* __restrict__ h, const float* __restrict__ Wh,
                            const float* __restrict__ bh, float* __restrict__ out, int n) {
  int i = blockIdx.x * blockDim.x + threadIdx.x;
  if (i >= n) return;
  const float* hp = h + (size_t)i * HDIM;
  float s = bh[0];
  #pragma unroll
  for (int k = 0; k < HDIM; ++k) s += hp[k] * Wh[k];
  out[i] = s;
}

// ---------------------------------------------------------------------------
extern "C" void kernel_launch(void* const* d_in, const int* in_sizes, int n_in,
                              void* d_out, int out_size, void* d_ws, size_t ws_size,
                              hipStream_t stream) {
  const float* x      = (const float*)d_in[0];
  const int*   eidx   = (const int*)d_in[1];
  const float* W_pre  = (const float*)d_in[2];
  const float* b_pre  = (const float*)d_in[3];
  const float* W1     = (const float*)d_in[4];
  const float* b1     = (const float*)d_in[5];
  const float* W2     = (const float*)d_in[6];
  const float* b2     = (const float*)d_in[7];
  const float* W_head = (const float*)d_in[8];
  const float* b_head = (const float*)d_in[9];
  float* out = (float*)d_out;

  const int N = in_sizes[0] / 128;   // 100000
  const int E = in_sizes[1] / 2;     // 1600000
  const int* src = eidx;
  const int* dst = eidx + E;

  // workspace: dis[N] | h[N*64] | hw[N*64] | acc[N*64]  (~77.2 MB)
  float* dis = (float*)d_ws;
  float* h   = dis + N;
  float* hw  = h   + (size_t)N * HDIM;
  float* acc = hw  + (size_t)N * HDIM;

  const int n64 = N * HDIM;

  // degree -> deg^-1/2
  zero_f32<<<(N + 255) / 256, 256, 0, stream>>>(dis, N);
  deg_accum<<<(E + 255) / 256, 256, 0, stream>>>(dst, dis, E);
  deg_to_dis<<<(N + 255) / 256, 256, 0, stream>>>(dis, N);

  // pre-MP: h = relu(x @ W_pre + b_pre)   (WMMA, K=128)
  int gemm_blocks = (N + 127) / 128;       // 8 waves/block * 16 rows/wave
  gemm_n64_wmma<128, true><<<gemm_blocks, 256, 0, stream>>>(x, W_pre, b_pre, h, N);

  const float* Ws[2] = {W1, W2};
  const float* bs[2] = {b1, b2};
  for (int layer = 0; layer < 2; ++layer) {
    // hw = h @ W   (WMMA, K=64)
    gemm_n64_wmma<64, false><<<gemm_blocks, 256, 0, stream>>>(h, Ws[layer], nullptr, hw, N);
    // acc = segment_sum(coef * hw[src], dst)
    zero_f32<<<(n64 + 255) / 256, 256, 0, stream>>>(acc, n64);
    long long sthreads = (long long)E * 32;
    scatter_edges<<<(int)((sthreads + 255) / 256), 256, 0, stream>>>(src, dst, dis, hw, acc, E);
    // h = relu(acc + b) + h
    finalize_layer<<<(n64 + 255) / 256, 256, 0, stream>>>(acc, bs[layer], h, n64);
  }

  // out = h @ W_head + b_head
  head_kernel<<<(N + 255) / 256, 256, 0, stream>>>(h, W_head, b_head, out, N);
}